// SVDppRecommender_49108656062664
// MI455X (gfx1250) — compile-verified
//
#include <hip/hip_runtime.h>

typedef __attribute__((ext_vector_type(2))) float v2f;
typedef __attribute__((ext_vector_type(8))) float v8f;

#define HIST 200
#define MU_CONST 3.5f
#define WAVES_PER_BLOCK 8

// One wave32 per batch element. Gather + sum 200 yj rows (float2 per lane),
// fuse user/item/W elementwise product, reduce across the wave with a
// V_WMMA_F32_16X16X4_F32 against an all-ones B matrix.
__global__ __launch_bounds__(256, 4) void svdpp_rating_kernel(
    const int* __restrict__ user_idx,
    const int* __restrict__ item_idx,
    const int* __restrict__ Iu,
    const float* __restrict__ user_emb,
    const float* __restrict__ item_emb,
    const float* __restrict__ yj,
    const float* __restrict__ user_bias,
    const float* __restrict__ item_bias,
    const float* __restrict__ W,
    const float* __restrict__ bscalar,
    float* __restrict__ out)
{
    const int lane = threadIdx.x & 31;
    const int wave = threadIdx.x >> 5;
    const int b    = blockIdx.x * WAVES_PER_BLOCK + wave;   // grid sized exactly

    // Wave-uniform user/item ids -> force scalar registers.
    const int uidx = __builtin_amdgcn_readfirstlane(user_idx[b]);
    const int iidx = __builtin_amdgcn_readfirstlane(item_idx[b]);

    // Preload the 200 wave-uniform history indices, lane-striped (7 coalesced loads).
    int hreg[7];
#pragma unroll
    for (int r = 0; r < 7; ++r) {
        const int h = r * 32 + lane;
        hreg[r] = (h < HIST) ? Iu[(size_t)b * HIST + h] : 0;
    }

    const v2f* __restrict__ yj2 = (const v2f*)yj;

    // 4 rotating accumulators -> long independent-load chains for MLP.
    v2f acc0 = {0.f, 0.f}, acc1 = {0.f, 0.f}, acc2 = {0.f, 0.f}, acc3 = {0.f, 0.f};
#pragma unroll
    for (int r = 0; r < 7; ++r) {
#pragma unroll
        for (int j = 0; j < 32; ++j) {
            if (r * 32 + j < HIST) {
                // Broadcast lane j's index -> SGPR; load becomes saddr + lane offset.
                const int sidx = __builtin_amdgcn_readlane(hreg[r], j);
                const v2f val = yj2[(size_t)sidx * 32 + lane];
                switch (j & 3) {
                    case 0: acc0 += val; break;
                    case 1: acc1 += val; break;
                    case 2: acc2 += val; break;
                    default: acc3 += val; break;
                }
            }
        }
    }
    const v2f acc = (acc0 + acc1) + (acc2 + acc3);

    const v2f uvec = ((const v2f*)user_emb)[(size_t)uidx * 32 + lane];
    const v2f ivec = ((const v2f*)item_emb)[(size_t)iidx * 32 + lane];
    const v2f wv   = ((const v2f*)W)[lane];

    const float inv_sqrt_hist = 0.07071067811865475f;  // 1/sqrt(200)
    float px = (uvec.x + acc.x * inv_sqrt_hist) * ivec.x * wv.x;
    float py = (uvec.y + acc.y * inv_sqrt_hist) * ivec.y * wv.y;
    float s  = px + py;  // per-lane partial of the 64-wide dot product

    // Wave-wide reduction via WMMA: A[m,0]=s(lane m), A[m,2]=s(lane m+16), A[*,1]=A[*,3]=0,
    // B = all-ones 4x16  =>  D[m,n] = s[m] + s[m+16] for every n.
    v2f A;  A.x = s;    A.y = 0.0f;
    v2f Bm; Bm.x = 1.0f; Bm.y = 1.0f;
    v8f C = {};
    v8f D = __builtin_amdgcn_wmma_f32_16x16x4_f32(
        /*neg_a=*/false, A, /*neg_b=*/false, Bm,
        /*c_mod=*/(short)0, C, /*reuse_a=*/false, /*reuse_b=*/false);

    // Sum rows within each half-wave (D vgpr v holds row v in lanes 0-15, row v+8 in 16-31)...
    float t = ((D[0] + D[1]) + (D[2] + D[3])) + ((D[4] + D[5]) + (D[6] + D[7]));
    // ...then combine the two half-wave row groups.
    float total = t + __shfl_xor(t, 16, 32);

    if (lane == 0) {
        out[b] = total + MU_CONST + bscalar[0] + user_bias[uidx] + item_bias[iidx];
    }
}

extern "C" void kernel_launch(void* const* d_in, const int* in_sizes, int n_in,
                              void* d_out, int out_size, void* d_ws, size_t ws_size,
                              hipStream_t stream)
{
    const int*   user_idx  = (const int*)d_in[0];
    const int*   item_idx  = (const int*)d_in[1];
    const int*   Iu        = (const int*)d_in[2];
    const float* user_emb  = (const float*)d_in[3];
    const float* item_emb  = (const float*)d_in[4];
    const float* yj        = (const float*)d_in[5];
    const float* user_bias = (const float*)d_in[6];
    const float* item_bias = (const float*)d_in[7];
    const float* W         = (const float*)d_in[8];
    const float* bscalar   = (const float*)d_in[9];
    float*       out       = (float*)d_out;

    const int batch = in_sizes[0];                 // 4096
    dim3 block(32 * WAVES_PER_BLOCK);              // 256 threads = 8 waves
    dim3 grid((batch + WAVES_PER_BLOCK - 1) / WAVES_PER_BLOCK);

    svdpp_rating_kernel<<<grid, block, 0, stream>>>(
        user_idx, item_idx, Iu, user_emb, item_emb, yj,
        user_bias, item_bias, W, bscalar, out);
}